// RandomProjectionQuantizerV2_28243704938614
// MI455X (gfx1250) — compile-verified
//
#include <hip/hip_runtime.h>

#define DIM 1024
#define CB_SIZE 8192
#define CB_DIM 16
#define NUM_CB 4
#define BATCH 4
#define SEQ 1024
#define LN_EPS 1e-5f
#define COS_EPS 1e-8f

typedef __attribute__((ext_vector_type(2))) float v2f;
typedef __attribute__((ext_vector_type(8))) float v8f;

// ---------------------------------------------------------------------------
// Kernel A: fused LayerNorm + random projection + row L2-normalize.
// One wave32 per (h, b, t). Output proj_n laid out (h,b,t,16) f32.
// LN folded: xn = (x-mean)*rstd  =>  proj_k = rstd*(dot(x,P_k) - mean*sum(P_k))
// ---------------------------------------------------------------------------
__global__ void __launch_bounds__(256) proj_kernel(const float* __restrict__ x,
                                                   const float* __restrict__ P,
                                                   float* __restrict__ proj_n) {
  const int lane = threadIdx.x & 31;
  const int wave = (int)((blockIdx.x * blockDim.x + threadIdx.x) >> 5); // h<<12 | b<<10 | t
  const int t = wave & (SEQ - 1);
  const int b = (wave >> 10) & (BATCH - 1);
  const int h = wave >> 12;
  const int half = lane >> 4;
  const int kk = lane & 15;

  const float* xrow = x + ((size_t)b * SEQ + t) * DIM;

  // mean / var over 1024 elements (full-wave reduction)
  float s = 0.f, s2 = 0.f;
  for (int d = lane; d < DIM; d += 32) {
    float v = xrow[d];
    s += v;
    s2 = fmaf(v, v, s2);
  }
  for (int off = 16; off > 0; off >>= 1) {
    s  += __shfl_xor(s,  off, 32);
    s2 += __shfl_xor(s2, off, 32);
  }
  const float mean = s * (1.f / DIM);
  const float var  = s2 * (1.f / DIM) - mean * mean;
  const float rstd = rsqrtf(var + LN_EPS);

  // dot products: lane handles output column kk, halves split the d-range
  const float* Ph = P + (size_t)h * DIM * CB_DIM;
  float s1 = 0.f, sp = 0.f;
  const int d0 = half * (DIM / 2);
  for (int d = d0; d < d0 + DIM / 2; ++d) {
    const float xv = xrow[d];
    const float pv = Ph[(size_t)d * CB_DIM + kk];   // 16 lanes -> 64B coalesced
    s1 = fmaf(xv, pv, s1);
    sp += pv;
  }
  s1 += __shfl_xor(s1, 16, 32);
  sp += __shfl_xor(sp, 16, 32);
  float val = rstd * (s1 - mean * sp);

  // L2 normalize over the 16 k-columns (reduction within 16-lane group)
  float nsq = val * val;
  for (int off = 8; off > 0; off >>= 1) nsq += __shfl_xor(nsq, off, 32);
  const float nrm = sqrtf(nsq);
  const float pn = val / fmaxf(nrm, COS_EPS);

  if (half == 0) proj_n[(size_t)wave * CB_DIM + kk] = pn;
}

// ---------------------------------------------------------------------------
// Kernel C: renormalize codebook with the reference's eps clamp.
// One thread per codebook row.
// ---------------------------------------------------------------------------
__global__ void __launch_bounds__(256) cbnorm_kernel(const float* __restrict__ CB,
                                                     float* __restrict__ cbn) {
  const int row = blockIdx.x * blockDim.x + threadIdx.x;   // 0 .. NUM_CB*CB_SIZE-1
  const float* src = CB + (size_t)row * CB_DIM;
  float v[CB_DIM];
  float ss = 0.f;
#pragma unroll
  for (int k = 0; k < CB_DIM; ++k) { v[k] = src[k]; ss = fmaf(v[k], v[k], ss); }
  const float inv = 1.f / fmaxf(sqrtf(ss), COS_EPS);
  float* dst = cbn + (size_t)row * CB_DIM;
#pragma unroll
  for (int k = 0; k < CB_DIM; ++k) dst[k] = v[k] * inv;
}

// ---------------------------------------------------------------------------
// Kernel B: distances GEMM (M=1024 t, N=8192 c, K=16) per (h,b) via
// V_WMMA_F32_16X16X4_F32 (4 chained WMMAs = K16), fused argmax, store tile.
// One wave per (hb, t-tile); 1024 waves total, each streams 512 c-tiles.
//
// A 16x4 f32 fragment (ISA layout): lanes 0-15 hold {K=0,K=1}, lanes 16-31
// hold {K=2,K=3} for rows M=lane&15.  B mirrored with N=lane&15 columns.
// C/D 16x16 f32: VGPR r holds M=r (lanes 0-15) / M=r+8 (lanes 16-31), N=lane&15.
// ---------------------------------------------------------------------------
__global__ void __launch_bounds__(256) dist_kernel(const float* __restrict__ proj_n,
                                                   const float* __restrict__ cbn,
                                                   float* __restrict__ dist,
                                                   int* __restrict__ indices) {
  const int lane  = threadIdx.x & 31;
  const int wave  = (int)((blockIdx.x * blockDim.x + threadIdx.x) >> 5); // 0..1023
  const int ttile = wave & 63;
  const int hb    = wave >> 6;          // h*4 + b
  const int h     = hb >> 2;
  const int b     = hb & 3;

  const int lo = lane & 15;
  const int hi = lane >> 4;
  const int t_base = ttile * 16;

  // Resident A fragments: proj_n rows t_base+lo, K slices {0..3},{4..7},{8..11},{12..15}
  const float* Arow = proj_n + (((size_t)hb * SEQ) + t_base + lo) * CB_DIM + hi * 2;
  const v2f a0 = *(const v2f*)(Arow + 0);
  const v2f a1 = *(const v2f*)(Arow + 4);
  const v2f a2 = *(const v2f*)(Arow + 8);
  const v2f a3 = *(const v2f*)(Arow + 12);

  float maxv[8];
  int   maxi[8];
#pragma unroll
  for (int r = 0; r < 8; ++r) { maxv[r] = -3.402823466e38f; maxi[r] = 0; }

  const float* CBh  = cbn + (size_t)h * CB_SIZE * CB_DIM;
  float*       drow = dist + (((size_t)hb * SEQ) + t_base) * CB_SIZE;

  for (int c_base = 0; c_base < CB_SIZE; c_base += 16) {
    // B fragments: codebook rows (= output columns) c_base+lo
    const float* Brow = CBh + (size_t)(c_base + lo) * CB_DIM + hi * 2;
    const v2f b0 = *(const v2f*)(Brow + 0);
    const v2f b1 = *(const v2f*)(Brow + 4);
    const v2f b2 = *(const v2f*)(Brow + 8);
    const v2f b3 = *(const v2f*)(Brow + 12);

    v8f acc = {};
    acc = __builtin_amdgcn_wmma_f32_16x16x4_f32(false, a0, false, b0, (short)0, acc, false, false);
    acc = __builtin_amdgcn_wmma_f32_16x16x4_f32(false, a1, false, b1, (short)0, acc, false, false);
    acc = __builtin_amdgcn_wmma_f32_16x16x4_f32(false, a2, false, b2, (short)0, acc, false, false);
    acc = __builtin_amdgcn_wmma_f32_16x16x4_f32(false, a3, false, b3, (short)0, acc, false, false);

    const int col = c_base + lo;
#pragma unroll
    for (int r = 0; r < 8; ++r) {
      const float d = acc[r];
      const int row = r + hi * 8;
      drow[(size_t)row * CB_SIZE + col] = d;
      if (d > maxv[r]) { maxv[r] = d; maxi[r] = col; }   // strict > keeps first index
    }
  }

  // argmax reduction across the 16 lanes of each half-group; lower index wins ties
#pragma unroll
  for (int r = 0; r < 8; ++r) {
    float mv = maxv[r];
    int   mi = maxi[r];
#pragma unroll
    for (int off = 8; off > 0; off >>= 1) {
      const float ov = __shfl_xor(mv, off, 32);
      const int   oi = __shfl_xor(mi, off, 32);
      if (ov > mv || (ov == mv && oi < mi)) { mv = ov; mi = oi; }
    }
    if (lo == 0) {
      const int trow = t_base + r + hi * 8;
      indices[((size_t)b * SEQ + trow) * NUM_CB + h] = mi;   // (b,t,h)
    }
  }
}

// ---------------------------------------------------------------------------
extern "C" void kernel_launch(void* const* d_in, const int* in_sizes, int n_in,
                              void* d_out, int out_size, void* d_ws, size_t ws_size,
                              hipStream_t stream) {
  const float* x  = (const float*)d_in[0];   // (4,1024,1024)
  const float* P  = (const float*)d_in[1];   // (4,1024,16)
  const float* CB = (const float*)d_in[2];   // (4,8192,16)

  // output tuple flat: indices (b,t,h) int32 then distances (h,b,t,c) f32
  int*   idx_out  = (int*)d_out;
  float* dist_out = (float*)d_out + (size_t)BATCH * SEQ * NUM_CB;

  // workspace: proj_n (1 MB) then cb_n (2 MB)
  float* proj_n = (float*)d_ws;
  float* cb_n   = proj_n + (size_t)NUM_CB * BATCH * SEQ * CB_DIM;

  // A: 16384 waves = 2048 blocks of 8 waves
  proj_kernel<<<2048, 256, 0, stream>>>(x, P, proj_n);
  // C: 32768 rows
  cbnorm_kernel<<<(NUM_CB * CB_SIZE) / 256, 256, 0, stream>>>(CB, cb_n);
  // B: 1024 waves = 128 blocks of 8 waves
  dist_kernel<<<128, 256, 0, stream>>>(proj_n, cb_n, dist_out, idx_out);
}